// SpatialGCN_5755256177393
// MI455X (gfx1250) — compile-verified
//
#include <hip/hip_runtime.h>
#include <hip/hip_bf16.h>

typedef __attribute__((ext_vector_type(2))) float v2f;
typedef __attribute__((ext_vector_type(8))) float v8f;

// ---------------------------------------------------------------------------
// Degree / normalization kernels
// ---------------------------------------------------------------------------
__global__ __launch_bounds__(256) void init_ones_kernel(float* __restrict__ p, int n) {
    int t = blockIdx.x * blockDim.x + threadIdx.x;
    if (t < n) p[t] = 1.0f;   // self-loop contributes degree 1
}

__global__ __launch_bounds__(256) void deg_accum_kernel(const int* __restrict__ dst,
                                                        float* __restrict__ deg, int e) {
    int t = blockIdx.x * blockDim.x + threadIdx.x;
    if (t < e) atomicAdd(&deg[dst[t]], 1.0f);
}

__global__ __launch_bounds__(256) void rsqrt_inplace_kernel(float* __restrict__ p, int n) {
    int t = blockIdx.x * blockDim.x + threadIdx.x;
    if (t < n) p[t] = rsqrtf(p[t]);
}

// ---------------------------------------------------------------------------
// WMMA GEMM:  hs[i,:] = dinv[i] * (X[i,:] @ W)   (K fixed = 128)
// Also writes acc_out = hs (self-loop initialization of the accumulator).
// One wave computes one 16x16 output tile; waves within a block cover OUTC.
// ---------------------------------------------------------------------------
template <int OUTC>
__global__ __launch_bounds__(32 * (OUTC / 16)) void gemm_scale_kernel(
    const float* __restrict__ X,     // [N,128]
    const float* __restrict__ W,     // [128,OUTC]
    const float* __restrict__ dinv,  // [N]
    float* __restrict__ hs,          // [N,OUTC]
    float* __restrict__ acc_out,     // [N,OUTC]
    int nRows) {
    constexpr int K = 128;
    constexpr int PAD = 132;                       // +4 pad: bank-conflict-free A frags
    constexpr int NTHR = 32 * (OUTC / 16);
    __shared__ float sX[16 * PAD];

    const int row0 = blockIdx.x * 16;
    const int tid = threadIdx.x;
    const bool fullTile = (row0 + 16 <= nRows);    // uniform across the block

    // Cooperative, coalesced load of the 16x128 X tile into LDS.
    if (fullTile) {
        for (int idx = tid; idx < 16 * (K / 4); idx += NTHR) {
            int r = idx >> 5;             // row in tile (K/4 == 32 float4 per row)
            int c4 = idx & 31;
            float4 v = ((const float4*)(X + (size_t)(row0 + r) * K))[c4];
            float* d = &sX[r * PAD + c4 * 4];
            d[0] = v.x; d[1] = v.y; d[2] = v.z; d[3] = v.w;
        }
    } else {
        for (int idx = tid; idx < 16 * (K / 4); idx += NTHR) {
            int r = idx >> 5;
            int c4 = idx & 31;
            int gr = row0 + r;
            if (gr >= nRows) gr = nRows - 1;
            float4 v = ((const float4*)(X + (size_t)gr * K))[c4];
            float* d = &sX[r * PAD + c4 * 4];
            d[0] = v.x; d[1] = v.y; d[2] = v.z; d[3] = v.w;
        }
    }
    __syncthreads();

    const int wave = tid >> 5;
    const int lane = tid & 31;
    const int col0 = wave * 16;
    const int mrow = lane & 15;          // A: lane -> M
    const int khalf = (lane >> 4) * 2;   // A/B: lanes 16..31 hold K+2,K+3

    v8f acc = {};
    const float* wp = W + col0 + mrow;   // B: lane -> N (column)

#pragma unroll 4
    for (int k = 0; k < K; k += 4) {
        v2f a;
        a.x = sX[mrow * PAD + k + khalf];
        a.y = sX[mrow * PAD + k + khalf + 1];
        v2f b;
        b.x = wp[(k + khalf) * OUTC];
        b.y = wp[(k + khalf + 1) * OUTC];
        acc = __builtin_amdgcn_wmma_f32_16x16x4_f32(false, a, false, b,
                                                    (short)0, acc, false, false);
    }

    // D layout: VGPR r holds D[M = r + 8*(lane>>4)][N = lane&15]
    const int nIdx = lane & 15;
    const int mhi = (lane >> 4) * 8;
    const int rowBase = row0 + mhi;
    if (fullTile) {
        // Straight-line stores: one base offset, 8 constant-stride rows.
        size_t base = (size_t)rowBase * OUTC + col0 + nIdx;
        float* __restrict__ h0 = hs + base;
        float* __restrict__ a0 = acc_out + base;
        const float* __restrict__ dv = dinv + rowBase;
#pragma unroll
        for (int r = 0; r < 8; ++r) {
            float val = acc[r] * dv[r];
            h0[(size_t)r * OUTC] = val;
            a0[(size_t)r * OUTC] = val;
        }
    } else {
#pragma unroll
        for (int r = 0; r < 8; ++r) {
            int grow = rowBase + r;
            if (grow < nRows) {
                float val = acc[r] * dinv[grow];
                size_t off = (size_t)grow * OUTC + col0 + nIdx;
                hs[off] = val;
                acc_out[off] = val;
            }
        }
    }
}

// ---------------------------------------------------------------------------
// Edge scatter: acc[dst,:] += hs[src,:]  (float4 gather, scalar f32 atomics)
// ---------------------------------------------------------------------------
template <int F>
__global__ __launch_bounds__(256) void edge_scatter_kernel(
    const int* __restrict__ src, const int* __restrict__ dst,
    const float* __restrict__ hs, float* __restrict__ acc, int nEdges) {
    constexpr int LPE = F / 4;                     // lanes per edge
    int tid = blockIdx.x * blockDim.x + threadIdx.x;
    int e = tid / LPE;
    if (e >= nEdges) return;
    int r4 = (tid % LPE) * 4;
    int s = src[e];
    int d = dst[e];
    const float4 v = *(const float4*)(hs + (size_t)s * F + r4);
    float* p = acc + (size_t)d * F + r4;
    atomicAdd(p + 0, v.x);
    atomicAdd(p + 1, v.y);
    atomicAdd(p + 2, v.z);
    atomicAdd(p + 3, v.w);
}

// ---------------------------------------------------------------------------
// Layer-1 finalize: z = relu(dinv[i]*acc + b)   (float4, F=128)
// ---------------------------------------------------------------------------
__global__ __launch_bounds__(256) void finalize_relu_kernel(
    const float* __restrict__ acc, const float* __restrict__ dinv,
    const float* __restrict__ b, float* __restrict__ z, int nQuads) {
    int tid = blockIdx.x * blockDim.x + threadIdx.x;
    if (tid >= nQuads) return;
    int i = tid >> 5;
    int c = (tid & 31) * 4;
    float s = dinv[i];
    float4 v = *(const float4*)(acc + (size_t)i * 128 + c);
    float4 bb = *(const float4*)(b + c);
    float4 o;
    o.x = fmaxf(fmaf(v.x, s, bb.x), 0.0f);
    o.y = fmaxf(fmaf(v.y, s, bb.y), 0.0f);
    o.z = fmaxf(fmaf(v.z, s, bb.z), 0.0f);
    o.w = fmaxf(fmaf(v.w, s, bb.w), 0.0f);
    *(float4*)(z + (size_t)i * 128 + c) = o;
}

// ---------------------------------------------------------------------------
// Layer-2 finalize + log_softmax over 64 classes, one wave32 per row, in place.
// ---------------------------------------------------------------------------
__global__ __launch_bounds__(256) void finalize_logsoftmax_kernel(
    float* __restrict__ out, const float* __restrict__ dinv,
    const float* __restrict__ b2, int nRows) {
    int row = blockIdx.x * 8 + (threadIdx.x >> 5);
    if (row >= nRows) return;
    int lane = threadIdx.x & 31;
    float s = dinv[row];
    float* p = out + (size_t)row * 64;
    float v0 = fmaf(p[lane], s, b2[lane]);
    float v1 = fmaf(p[lane + 32], s, b2[lane + 32]);
    float m = fmaxf(v0, v1);
#pragma unroll
    for (int off = 16; off > 0; off >>= 1) m = fmaxf(m, __shfl_xor(m, off, 32));
    float sum = expf(v0 - m) + expf(v1 - m);
#pragma unroll
    for (int off = 16; off > 0; off >>= 1) sum += __shfl_xor(sum, off, 32);
    float lse = logf(sum) + m;
    p[lane] = v0 - lse;
    p[lane + 32] = v1 - lse;
}

// ---------------------------------------------------------------------------
// Host-side launch
// ---------------------------------------------------------------------------
extern "C" void kernel_launch(void* const* d_in, const int* in_sizes, int n_in,
                              void* d_out, int out_size, void* d_ws, size_t ws_size,
                              hipStream_t stream) {
    (void)n_in; (void)out_size; (void)ws_size;
    const float* x  = (const float*)d_in[0];
    const int*   ei = (const int*)d_in[1];
    const float* W1 = (const float*)d_in[2];
    const float* b1 = (const float*)d_in[3];
    const float* W2 = (const float*)d_in[4];
    const float* b2 = (const float*)d_in[5];
    float* out = (float*)d_out;

    const int FIN = 128, HID = 128;
    const int N = in_sizes[0] / FIN;
    const int E = in_sizes[1] / 2;
    const int* srcI = ei;
    const int* dstI = ei + E;

    // Workspace layout (floats): dinv | hs1[N*128] | acc1[N*128] | hs2[N*64]
    size_t dinvElems = ((size_t)N + 1023) & ~(size_t)1023;
    float* dinv = (float*)d_ws;
    float* hs1  = dinv + dinvElems;
    float* acc1 = hs1 + (size_t)N * HID;
    float* hs2  = acc1 + (size_t)N * HID;

    const int B = 256;
    // 1) degrees + dinv
    init_ones_kernel<<<(N + B - 1) / B, B, 0, stream>>>(dinv, N);
    deg_accum_kernel<<<(E + B - 1) / B, B, 0, stream>>>(dstI, dinv, E);
    rsqrt_inplace_kernel<<<(N + B - 1) / B, B, 0, stream>>>(dinv, N);

    // 2) layer 1: hs1 = dinv * (x @ W1); acc1 initialized with self-loop term
    int rowTiles = (N + 15) / 16;
    gemm_scale_kernel<128><<<rowTiles, 256, 0, stream>>>(x, W1, dinv, hs1, acc1, N);

    // 3) scatter-add over edges (128 features)
    long long th1 = (long long)E * 32;
    edge_scatter_kernel<128><<<(unsigned)((th1 + B - 1) / B), B, 0, stream>>>(
        srcI, dstI, hs1, acc1, E);

    // 4) z1 = relu(dinv*acc1 + b1)  (reuse hs1 buffer as z1)
    int nQuads = N * 32;
    finalize_relu_kernel<<<(nQuads + B - 1) / B, B, 0, stream>>>(acc1, dinv, b1, hs1, nQuads);

    // 5) layer 2: hs2 = dinv * (z1 @ W2); d_out used as accumulator
    gemm_scale_kernel<64><<<rowTiles, 128, 0, stream>>>(hs1, W2, dinv, hs2, out, N);

    // 6) scatter-add over edges (64 features)
    long long th2 = (long long)E * 16;
    edge_scatter_kernel<64><<<(unsigned)((th2 + B - 1) / B), B, 0, stream>>>(
        srcI, dstI, hs2, out, E);

    // 7) bias + log_softmax in place
    finalize_logsoftmax_kernel<<<(N + 7) / 8, 256, 0, stream>>>(out, dinv, b2, N);
}